// Repro_23278722744950
// MI455X (gfx1250) — compile-verified
//
#include <hip/hip_runtime.h>
#include <stdint.h>

// SwiGLU + per-128-group dynamic fp8(e4m3) quantization for MI455X (gfx1250).
// Memory-bound streaming kernel: wave32, one wave per token, async global->LDS
// double-buffered staging, hardware v_cvt_pk_fp8_f32 packing.

typedef __attribute__((ext_vector_type(8))) _Float16 v8h;   // 16 bytes
typedef int v4i __attribute__((ext_vector_type(4)));        // 16 bytes

#define WAVE            32
#define WAVES_PER_BLOCK 8
#define BLOCK           (WAVE * WAVES_PER_BLOCK)
#define ROW_IN          512          // fp16 per token (gate|up)
#define ROW_H           256          // h elements per token
#define ROW_BYTES       1024         // ROW_IN * 2 bytes

#if defined(__gfx1250__) && \
    __has_builtin(__builtin_amdgcn_global_load_async_to_lds_b128) && \
    __has_builtin(__builtin_amdgcn_s_wait_asynccnt)
#define HAVE_ASYNC 1
#else
#define HAVE_ASYNC 0
#endif

#if HAVE_ASYNC
typedef __attribute__((address_space(1))) v4i gv4i;   // global int4
typedef __attribute__((address_space(3))) v4i lv4i;   // LDS int4
__device__ __forceinline__ gv4i* to_global(const void* p) {
  return (gv4i*)(uintptr_t)p;
}
__device__ __forceinline__ lv4i* to_lds(const void* p) {
  // generic->LDS: low 32 bits of a generic LDS pointer are the LDS offset
  return (lv4i*)(uint32_t)(uintptr_t)p;
}
#endif

// ---------------- fp8 e4m3 (OCP) packing ----------------

__device__ __forceinline__ unsigned f32_to_e4m3_sw(float x) {
  unsigned u   = __float_as_uint(x);
  unsigned sgn = (u >> 24) & 0x80u;
  unsigned au  = u & 0x7FFFFFFFu;
  if (au >= 0x43E00000u) return sgn | 0x7Eu;            // |x| >= 448 -> max
  int e = (int)(au >> 23) - 127;
  if (e < -6) {                                         // denormal target
    int mi = (int)rintf(__uint_as_float(au) * 512.0f);  // x * 2^9, RNE
    return sgn | (unsigned)mi;                          // mi in [0,8]
  }
  unsigned r  = au + 0x7FFFFu + ((au >> 20) & 1u);      // RNE, drop 20 bits
  unsigned m3 = (r >> 20) & 7u;
  int eb = (int)(r >> 23) - 127 + 7;
  if (eb >= 16) return sgn | 0x7Eu;
  return sgn | ((unsigned)eb << 3) | m3;
}

__device__ __forceinline__ uint32_t pack4_fp8(float a, float b, float c, float d) {
#if defined(__gfx1250__) && __has_builtin(__builtin_amdgcn_cvt_pk_fp8_f32)
  int r = 0;
  r = __builtin_amdgcn_cvt_pk_fp8_f32(a, b, r, false);  // word 0
  r = __builtin_amdgcn_cvt_pk_fp8_f32(c, d, r, true);   // word 1
  return (uint32_t)r;
#else
  return f32_to_e4m3_sw(a) | (f32_to_e4m3_sw(b) << 8) |
         (f32_to_e4m3_sw(c) << 16) | (f32_to_e4m3_sw(d) << 24);
#endif
}

// ---------------- per-token compute ----------------
// Lane owns h[8*lane .. 8*lane+7]; lanes 0-15 = group 0, lanes 16-31 = group 1.

__device__ __forceinline__ void process_token(v8h g8, v8h u8, int lane,
                                              uint8_t* __restrict__ qrow,
                                              float* __restrict__ srow) {
  float hf[8];
  float amax = 0.0f;
#pragma unroll
  for (int j = 0; j < 8; ++j) {
    float g   = (float)g8[j];
    float sig = 1.0f / (1.0f + __expf(-g));
    _Float16 hh = (_Float16)(g * sig) * u8[j];   // fp16 cast then fp16 mul (ref)
    float f = (float)hh;
    hf[j] = f;
    amax  = fmaxf(amax, fabsf(f));
  }
  // absmax across the 16-lane half-wave that owns this 128-group
#pragma unroll
  for (int m = 1; m < 16; m <<= 1)
    amax = fmaxf(amax, __shfl_xor(amax, m, 32));

  amax = fmaxf(amax, 1e-10f);
  float sc  = amax * (1.0f / 448.0f);
  float inv = 448.0f / amax;

  float v[8];
#pragma unroll
  for (int j = 0; j < 8; ++j)
    v[j] = fminf(fmaxf(hf[j] * inv, -448.0f), 448.0f);

  uint2 packed;
  packed.x = pack4_fp8(v[0], v[1], v[2], v[3]);
  packed.y = pack4_fp8(v[4], v[5], v[6], v[7]);
  *(uint2*)(qrow + lane * 8) = packed;           // global_store_b64

  if ((lane & 15) == 0) srow[lane >> 4] = sc;    // one scale per group
}

#if HAVE_ASYNC
__device__ __forceinline__ void prefetch_token(const _Float16* __restrict__ x,
                                               size_t t, uint8_t* dst, int lane) {
  const uint8_t* gsrc = (const uint8_t*)(x + t * ROW_IN) + lane * 16;
  // gate half -> [0,512), up half -> [512,1024) of the LDS buffer
  __builtin_amdgcn_global_load_async_to_lds_b128(to_global(gsrc),
                                                 to_lds(dst + lane * 16), 0, 0);
  __builtin_amdgcn_global_load_async_to_lds_b128(to_global(gsrc + 512),
                                                 to_lds(dst + 512 + lane * 16), 0, 0);
}
#endif

__global__ __launch_bounds__(BLOCK) void swiglu_fp8_kernel(
    const _Float16* __restrict__ x, uint8_t* __restrict__ q,
    float* __restrict__ scl, int T, int nwaves) {
  const int lane = threadIdx.x & (WAVE - 1);
  const int wid  = (int)((blockIdx.x * (unsigned)blockDim.x + threadIdx.x) >> 5);

#if HAVE_ASYNC
  __shared__ __align__(16) uint8_t smem[WAVES_PER_BLOCK * 2 * ROW_BYTES];
  uint8_t* buf = smem + (threadIdx.x >> 5) * (2 * ROW_BYTES);

  int t = wid;
  if (t < T) prefetch_token(x, (size_t)t, buf, lane);
  int b = 0;
  for (; t < T; t += nwaves, b ^= 1) {
    int nt = t + nwaves;
    if (nt < T) {
      prefetch_token(x, (size_t)nt, buf + (b ^ 1) * ROW_BYTES, lane);
      __builtin_amdgcn_s_wait_asynccnt(2);   // current token's 2 loads done
    } else {
      __builtin_amdgcn_s_wait_asynccnt(0);
    }
    asm volatile("" ::: "memory");           // keep LDS reads below the wait
    const uint8_t* cur = buf + b * ROW_BYTES;
    v8h g8 = *(const v8h*)(cur + lane * 16);         // ds_load_b128
    v8h u8 = *(const v8h*)(cur + 512 + lane * 16);   // ds_load_b128
    process_token(g8, u8, lane, q + (size_t)t * ROW_H, scl + (size_t)t * 2);
  }
#else
  for (int t = wid; t < T; t += nwaves) {
    const v8h* row = (const v8h*)(x + (size_t)t * ROW_IN);
    v8h g8 = row[lane];        // gate chunk: halves [8*lane, 8*lane+8)
    v8h u8 = row[32 + lane];   // up   chunk: halves [256+8*lane, ...)
    process_token(g8, u8, lane, q + (size_t)t * ROW_H, scl + (size_t)t * 2);
  }
#endif
}

extern "C" void kernel_launch(void* const* d_in, const int* in_sizes, int n_in,
                              void* d_out, int out_size, void* d_ws, size_t ws_size,
                              hipStream_t stream) {
  (void)n_in; (void)d_ws; (void)ws_size; (void)out_size;
  const _Float16* x = (const _Float16*)d_in[0];
  const int T = in_sizes[0] / ROW_IN;                 // 262144 tokens

  uint8_t* q  = (uint8_t*)d_out;                      // T*256 fp8 bytes
  float*  scl = (float*)(q + (size_t)T * ROW_H);      // then T*2 fp32 scales

  const int blocks = 4096;                            // 32768 waves, ~8 tok/wave
  const int nwaves = blocks * WAVES_PER_BLOCK;
  swiglu_fp8_kernel<<<dim3(blocks), dim3(BLOCK), 0, stream>>>(x, q, scl, T, nwaves);
}